// YOLOXLoss_37280316130066
// MI455X (gfx1250) — compile-verified
//
#include <hip/hip_runtime.h>
#include <math.h>

// ---------------------------------------------------------------------------
// YOLOX SimOTA loss on gfx1250 (MI455X).
// WMMA: cost_cls gather = onehot(gcls) @ (log2q - log2p) via
// V_WMMA_F32_16X16X4_F32 — exact (single nonzero product per K-dot).
// Cost matrix is ranking-only (stop_gradient in reference), so it uses raw
// TRANS ops (v_exp/v_log/v_rcp/v_sqrt) and is computed in log2 domain with a
// single *ln2 at the end.  Loss-bearing math (BCE, GIoU, dyn_k IoU) stays on
// precise libm to track the reference numerics.
// ---------------------------------------------------------------------------

typedef float v2f __attribute__((ext_vector_type(2)));
typedef float v8f __attribute__((ext_vector_type(8)));

constexpr int   B_  = 32;
constexpr int   N_  = 8400;
constexpr int   M_  = 32;
constexpr int   NC_ = 80;
constexpr int   PS_ = 85;          // prediction row stride (4 box + 1 obj + 80 cls)
constexpr float EPS_ = 1e-7f;
constexpr float LN2_ = 0.69314718055994530942f;

// precise helpers (loss-bearing)
__device__ __forceinline__ float bcef_(float x, float t) {
    return fmaxf(x, 0.0f) - x * t + log1pf(expf(-fabsf(x)));
}
// fast helpers (ranking-only): raw v_rcp_f32 / v_exp_f32 / v_log_f32 (log2)
__device__ __forceinline__ float frcp_(float x)   { return __builtin_amdgcn_rcpf(x); }
__device__ __forceinline__ float fsig_(float x)   { return frcp_(1.0f + __expf(-x)); }
__device__ __forceinline__ float flog2_(float x)  { return __builtin_amdgcn_logf(x); }
// monotonic float -> uint key (handles any sign exactly)
__device__ __forceinline__ unsigned fkey_(float f) {
    unsigned u = __float_as_uint(f);
    return (u & 0x80000000u) ? ~u : (u | 0x80000000u);
}

// ---------------------------------------------------------------------------
// Kernel 1: per-(image, 16-anchor tile) wave.
// A 16x4 f32 layout (documented): lane = M (0..15 both halves), VGPR j holds
// K = j + 2*(lane>=16).  B mirrors that K striping with lane&15 = N column.
// C/D: VGPR j, lane l -> M = j + 8*(l>>4), N = l&15.
// ---------------------------------------------------------------------------
__global__ __launch_bounds__(256) void yolox_cost_kernel(
    const float* __restrict__ pred, const float* __restrict__ tgt,
    float* __restrict__ cost, float4* __restrict__ pxy,
    unsigned* __restrict__ inbox)
{
    __shared__ float sgx1[M_], sgy1[M_], sgx2[M_], sgy2[M_], sga[M_];
    __shared__ int   sgc[M_];

    const int b   = blockIdx.x;
    const int tid = threadIdx.x;
    if (tid < M_) {
        const float* t = tgt + (size_t)(b * M_ + tid) * 6;
        sgc[tid] = (int)t[0];
        float cx = t[1], cy = t[2], w = t[3], h = t[4];
        float x1 = cx - 0.5f * w, y1 = cy - 0.5f * h;
        float x2 = cx + 0.5f * w, y2 = cy + 0.5f * h;
        sgx1[tid] = x1; sgy1[tid] = y1; sgx2[tid] = x2; sgy2[tid] = y2;
        sga[tid]  = fmaxf(x2 - x1, 0.0f) * fmaxf(y2 - y1, 0.0f);
    }
    __syncthreads();

    const int lane = tid & 31;
    const int wave = tid >> 5;
    const int tile = blockIdx.y * 8 + wave;
    if (tile * 16 >= N_) return;              // wave-uniform -> EXEC stays full for WMMA
    const int al = lane & 15;                 // anchor within tile (N column)
    const int h2 = lane >> 4;                 // lane half -> K offset
    const int a  = tile * 16 + al;

    const float* p = pred + (size_t)(b * N_ + a) * PS_;
    const float pcx = p[0], pcy = p[1], pw = p[2], ph = p[3], pobj = p[4];
    const float sobj = fsig_(pobj);           // ranking-only
    const float px1 = pcx - 0.5f * pw, py1 = pcy - 0.5f * ph;
    const float px2 = pcx + 0.5f * pw, py2 = pcy + 0.5f * ph;
    const float pa  = fmaxf(px2 - px1, 0.0f) * fmaxf(py2 - py1, 0.0f);
    if (h2 == 0) pxy[(size_t)b * N_ + a] = make_float4(px1, py1, px2, py2);

    const int gcA = sgc[al];        // one-hot class for gt tile 0 (gts 0..15)
    const int gcB = sgc[16 + al];   // gt tile 1 (gts 16..31)

    v8f acc0 = {0, 0, 0, 0, 0, 0, 0, 0};
    v8f acc1 = {0, 0, 0, 0, 0, 0, 0, 0};
    float base2 = 0.0f;                       // sum of log2(1-s+eps), log2 domain
    const float* pc = p + 5;

#pragma unroll 4
    for (int chunk = 0; chunk < 20; ++chunk) {
        const int c0 = chunk * 4 + 2 * h2;    // this lane's two classes: c0, c0+1
        float x0 = pc[c0], x1 = pc[c0 + 1];
        float s0 = __builtin_amdgcn_sqrtf(fsig_(x0) * sobj);
        float s1 = __builtin_amdgcn_sqrtf(fsig_(x1) * sobj);
        float lp0 = flog2_(s0 + EPS_), lq0 = flog2_(fmaxf(1.0f - s0, 0.0f) + EPS_);
        float lp1 = flog2_(s1 + EPS_), lq1 = flog2_(fmaxf(1.0f - s1, 0.0f) + EPS_);
        base2 += lq0 + lq1;                   // half of sum(lq); merged via shfl below
        v2f Bv; Bv.x = lq0 - lp0;           Bv.y = lq1 - lp1;
        v2f A0; A0.x = (gcA == c0) ? 1.0f : 0.0f; A0.y = (gcA == c0 + 1) ? 1.0f : 0.0f;
        v2f A1; A1.x = (gcB == c0) ? 1.0f : 0.0f; A1.y = (gcB == c0 + 1) ? 1.0f : 0.0f;
        acc0 = __builtin_amdgcn_wmma_f32_16x16x4_f32(false, A0, false, Bv, (short)0, acc0, false, false);
        acc1 = __builtin_amdgcn_wmma_f32_16x16x4_f32(false, A1, false, Bv, (short)0, acc1, false, false);
    }
    base2 += __shfl_xor(base2, 16, 32);       // combine the two class halves

    unsigned msk = 0;
#pragma unroll
    for (int j = 0; j < 8; ++j) {
#pragma unroll
        for (int t2 = 0; t2 < 2; ++t2) {
            const int g = t2 * 16 + j + 8 * h2;
            const float ccls2 = (t2 == 0) ? acc0[j] : acc1[j];   // (lq-lp)[gcls], log2
            float gx1 = sgx1[g], gy1 = sgy1[g], gx2 = sgx2[g], gy2 = sgy2[g];
            float ix1 = fmaxf(gx1, px1), iy1 = fmaxf(gy1, py1);
            float ix2 = fminf(gx2, px2), iy2 = fminf(gy2, py2);
            float inter = fmaxf(ix2 - ix1, 0.0f) * fmaxf(iy2 - iy1, 0.0f);
            float iou = inter * frcp_(sga[g] + pa - inter + EPS_);   // ranking-only
            bool inb = (pcx > gx1) && (pcx < gx2) && (pcy > gy1) && (pcy < gy2);
            // cost = ln2 * (cost_cls2 + 3*reg_cost2) + big*(!in_box)
            //   cost_cls2 = ccls2 - base2   (base = -sum(lq) in the reference)
            float c = LN2_ * (ccls2 - base2 - 3.0f * flog2_(iou + 1e-8f))
                      + (inb ? 0.0f : 100000.0f);
            cost[(size_t)(b * M_ + g) * N_ + a] = c;
            if (inb) msk |= (1u << g);
        }
    }
    msk |= __shfl_xor(msk, 16, 32);
    if (h2 == 0) inbox[(size_t)b * N_ + a] = msk;
}

// ---------------------------------------------------------------------------
// Kernel 2: one wave per (image, gt).  Single scan keeps per-lane top-10
// masked IoU (precise math: feeds dyn_k = int(sum)) and bottom-10
// (key(cost)<<32|anchor) keys.  Wave shfl k-way merges yield dyn_k and the
// dyn_k-th smallest key K*, reproducing argsort(argsort(cost)) rank semantics
// with stable index tie-break.
// ---------------------------------------------------------------------------
__global__ __launch_bounds__(256) void yolox_select_kernel(
    const float* __restrict__ tgt, const float* __restrict__ cost,
    const float4* __restrict__ pxy, const unsigned* __restrict__ inbox,
    unsigned long long* __restrict__ keystar)
{
    const int wglobal = blockIdx.x * 8 + (threadIdx.x >> 5);
    const int lane = threadIdx.x & 31;
    const int b = wglobal >> 5;
    const int g = wglobal & 31;

    const float* t = tgt + (size_t)(b * M_ + g) * 6;
    const float cx = t[1], cy = t[2], w = t[3], h = t[4];
    const float gx1 = cx - 0.5f * w, gy1 = cy - 0.5f * h;
    const float gx2 = cx + 0.5f * w, gy2 = cy + 0.5f * h;
    const float ga  = fmaxf(gx2 - gx1, 0.0f) * fmaxf(gy2 - gy1, 0.0f);

    const float*    crow = cost + (size_t)(b * M_ + g) * N_;
    const float4*   prow = pxy + (size_t)b * N_;
    const unsigned* mrow = inbox + (size_t)b * N_;

    float ti[10];
    unsigned long long ck[10];
#pragma unroll
    for (int i = 0; i < 10; ++i) { ti[i] = 0.0f; ck[i] = ~0ull; }

    for (int n = lane; n < N_; n += 32) {
        float4 pb = prow[n];
        float pa2 = fmaxf(pb.z - pb.x, 0.0f) * fmaxf(pb.w - pb.y, 0.0f);
        float ix1 = fmaxf(gx1, pb.x), iy1 = fmaxf(gy1, pb.y);
        float ix2 = fminf(gx2, pb.z), iy2 = fminf(gy2, pb.w);
        float inter = fmaxf(ix2 - ix1, 0.0f) * fmaxf(iy2 - iy1, 0.0f);
        float iou = inter / (ga + pa2 - inter + EPS_);       // precise: feeds dyn_k
        float ioum = ((mrow[n] >> g) & 1u) ? iou : 0.0f;
#pragma unroll
        for (int i = 0; i < 10; ++i) {            // keep descending top-10
            if (ioum > ti[i]) { float tmp = ti[i]; ti[i] = ioum; ioum = tmp; }
        }
        unsigned long long k =
            (((unsigned long long)fkey_(crow[n])) << 32) | (unsigned)n;
#pragma unroll
        for (int i = 0; i < 10; ++i) {            // keep ascending bottom-10
            if (k < ck[i]) { unsigned long long tmp = ck[i]; ck[i] = k; k = tmp; }
        }
    }

    // --- wave merge: top-10 IoU sum (iou >= 0 -> bit order is value order) ---
    float sum = 0.0f;
    int ptr = 0;
    for (int r = 0; r < 10; ++r) {
        unsigned long long key = (ptr < 10)
            ? ((((unsigned long long)__float_as_uint(ti[ptr])) << 5) | (unsigned)lane)
            : 0ull;
        unsigned long long mx = key;
#pragma unroll
        for (int s = 16; s >= 1; s >>= 1) {
            unsigned long long o = __shfl_xor(mx, s, 32);
            if (o > mx) mx = o;
        }
        sum += __uint_as_float((unsigned)(mx >> 5));
        if (key == mx) ++ptr;
    }
    int dynk = (int)sum;
    if (dynk < 1) dynk = 1;
    if (dynk > 10) dynk = 10;

    // --- wave merge: dyn_k-th smallest cost key ---
    int p2 = 0;
    unsigned long long kstar = 0ull;
    for (int r = 0; r < dynk; ++r) {              // dynk is wave-uniform
        unsigned long long key = (p2 < 10) ? ck[p2] : ~0ull;
        unsigned long long mn = key;
#pragma unroll
        for (int s = 16; s >= 1; s >>= 1) {
            unsigned long long o = __shfl_xor(mn, s, 32);
            if (o < mn) mn = o;
        }
        kstar = mn;
        if (key == mn) ++p2;
    }
    if (lane == 0) keystar[b * M_ + g] = kstar;
}

// ---------------------------------------------------------------------------
// Kernel 3: per anchor.  assigned(g) = inbox bit & (key <= K*[g]);
// matched_gt = first-min cost among assigned; precise BCE obj/cls + GIoU.
// Double partials per block -> deterministic final reduction.
// ---------------------------------------------------------------------------
__global__ __launch_bounds__(256) void yolox_final_kernel(
    const float* __restrict__ pred, const float* __restrict__ tgt,
    const float* __restrict__ cost, const float4* __restrict__ pxy,
    const unsigned* __restrict__ inbox,
    const unsigned long long* __restrict__ keystar,
    double* __restrict__ partials)
{
    __shared__ float sgx1[M_], sgy1[M_], sgx2[M_], sgy2[M_];
    __shared__ int   sgc[M_];
    __shared__ unsigned long long sks[M_];
    __shared__ double redl[256];
    __shared__ double redn[256];

    const int b = blockIdx.x;
    const int tid = threadIdx.x;
    if (tid < M_) {
        const float* t = tgt + (size_t)(b * M_ + tid) * 6;
        sgc[tid] = (int)t[0];
        float cx = t[1], cy = t[2], w = t[3], h = t[4];
        sgx1[tid] = cx - 0.5f * w; sgy1[tid] = cy - 0.5f * h;
        sgx2[tid] = cx + 0.5f * w; sgy2[tid] = cy + 0.5f * h;
        sks[tid] = keystar[b * M_ + tid];
    }
    __syncthreads();

    const int n = blockIdx.y * 256 + tid;
    double contrib = 0.0, nf = 0.0;
    if (n < N_) {
        const unsigned m = inbox[(size_t)b * N_ + n];
        const float* crow = cost + (size_t)b * M_ * N_ + n;
        bool fg = false;
        float best = 1e30f;
        int mg = 0;
#pragma unroll 4
        for (int g = 0; g < M_; ++g) {
            float c = crow[(size_t)g * N_];
            unsigned long long k =
                (((unsigned long long)fkey_(c)) << 32) | (unsigned)n;
            bool asg = ((m >> g) & 1u) && (k <= sks[g]);
            if (asg) {
                fg = true;
                if (c < best) { best = c; mg = g; }
            }
        }
        const float* p = pred + (size_t)(b * N_ + n) * PS_;
        float tot = bcef_(p[4], fg ? 1.0f : 0.0f);       // obj loss (all anchors)
        if (fg) {
            const int gc = sgc[mg];
            float cls_l = 0.0f;
            for (int c = 0; c < NC_; ++c)
                cls_l += bcef_(p[5 + c], (c == gc) ? 1.0f : 0.0f);
            float4 pb = pxy[(size_t)b * N_ + n];
            float tx1 = sgx1[mg], ty1 = sgy1[mg], tx2 = sgx2[mg], ty2 = sgy2[mg];
            float ix1 = fmaxf(pb.x, tx1), iy1 = fmaxf(pb.y, ty1);
            float ix2 = fminf(pb.z, tx2), iy2 = fminf(pb.w, ty2);
            float inter = fmaxf(ix2 - ix1, 0.0f) * fmaxf(iy2 - iy1, 0.0f);
            float pa = fmaxf(pb.z - pb.x, 0.0f) * fmaxf(pb.w - pb.y, 0.0f);
            float ta = fmaxf(tx2 - tx1, 0.0f) * fmaxf(ty2 - ty1, 0.0f);
            float uni = pa + ta - inter + EPS_;
            float iou = inter / uni;
            float ex1 = fminf(pb.x, tx1), ey1 = fminf(pb.y, ty1);
            float ex2 = fmaxf(pb.z, tx2), ey2 = fmaxf(pb.w, ty2);
            float enc = (ex2 - ex1) * (ey2 - ey1) + EPS_;
            float gio = iou - (enc - uni) / enc;
            tot += cls_l + 5.0f * (1.0f - gio);          // CLS_W=1, REG_W=5
            nf = 1.0;
        }
        contrib = (double)tot;
    }
    redl[tid] = contrib;
    redn[tid] = nf;
    __syncthreads();
#pragma unroll
    for (int s = 128; s > 0; s >>= 1) {
        if (tid < s) { redl[tid] += redl[tid + s]; redn[tid] += redn[tid + s]; }
        __syncthreads();
    }
    if (tid == 0) {
        int bid = b * gridDim.y + blockIdx.y;
        partials[2 * bid]     = redl[0];
        partials[2 * bid + 1] = redn[0];
    }
}

// Kernel 4: deterministic fixed-order final sum + normalization.
__global__ void yolox_reduce_kernel(const double* __restrict__ partials, int np,
                                    float* __restrict__ out)
{
    if (threadIdx.x == 0 && blockIdx.x == 0) {
        double s = 0.0, n = 0.0;
        for (int i = 0; i < np; ++i) { s += partials[2 * i]; n += partials[2 * i + 1]; }
        double denom = (n > 1.0) ? n : 1.0;
        out[0] = (float)(s / denom);
    }
}

extern "C" void kernel_launch(void* const* d_in, const int* in_sizes, int n_in,
                              void* d_out, int out_size, void* d_ws, size_t ws_size,
                              hipStream_t stream)
{
    const float* pred = (const float*)d_in[0];   // (B, N, 85) f32
    const float* tgt  = (const float*)d_in[1];   // (B, M, 6)  f32
    float* out = (float*)d_out;

    // Workspace layout (all regions fully rewritten every launch):
    char* ws = (char*)d_ws;
    size_t off = 0;
    float* cost = (float*)(ws + off);               off += (size_t)B_ * M_ * N_ * 4;  // 34.4 MB
    float4* pxy = (float4*)(ws + off);              off += (size_t)B_ * N_ * 16;      //  4.3 MB
    unsigned* inbox = (unsigned*)(ws + off);        off += (size_t)B_ * N_ * 4;       //  1.1 MB
    unsigned long long* ks = (unsigned long long*)(ws + off); off += (size_t)B_ * M_ * 8;
    double* part = (double*)(ws + off);

    const int TILES = N_ / 16;                      // 525
    const int NBY   = (N_ + 255) / 256;             // 33

    yolox_cost_kernel<<<dim3(B_, (TILES + 7) / 8), 256, 0, stream>>>(pred, tgt, cost, pxy, inbox);
    yolox_select_kernel<<<dim3((B_ * M_) / 8), 256, 0, stream>>>(tgt, cost, pxy, inbox, ks);
    yolox_final_kernel<<<dim3(B_, NBY), 256, 0, stream>>>(pred, tgt, cost, pxy, inbox, ks, part);
    yolox_reduce_kernel<<<1, 32, 0, stream>>>(part, B_ * NBY, out);
}